// KipfGCN_1743756722177
// MI455X (gfx1250) — compile-verified
//
#include <hip/hip_runtime.h>
#include <hip/hip_bf16.h>

// ---------------------------------------------------------------------------
// Kipf 2-layer GCN for MI455X (gfx1250, wave32).
//   out = log_softmax( A_hat @ relu(A_hat @ X @ W1 + b1) @ W2 + b2 )
// A_hat = D^-1/2 (A + I) D^-1/2 applied as gather/scale/atomic-scatter.
// Dense transforms: V_WMMA_F32_16X16X4_F32 (fp32-exact; GEMMs are only
// ~4 GFLOP vs ~1.5 GB of scatter traffic, so fp32 exactness is free).
// ---------------------------------------------------------------------------

#define NNODES 100000

typedef float v2f __attribute__((ext_vector_type(2)));
typedef float v4f __attribute__((ext_vector_type(4)));
typedef float v8f __attribute__((ext_vector_type(8)));

// ---------------- degree / norm ----------------

__global__ __launch_bounds__(256) void k_deg_init(float* __restrict__ deg) {
    int i = blockIdx.x * 256 + threadIdx.x;
    if (i < NNODES) deg[i] = 1.0f;            // self-loop contribution
}

__global__ __launch_bounds__(256) void k_deg_accum(const int* __restrict__ dst,
                                                   int E, float* __restrict__ deg) {
    int e = blockIdx.x * 256 + threadIdx.x;
    if (e < E) atomicAdd(&deg[dst[e]], 1.0f);
}

__global__ __launch_bounds__(256) void k_deg_to_dinv(float* __restrict__ d) {
    int i = blockIdx.x * 256 + threadIdx.x;
    if (i < NNODES) d[i] = rsqrtf(d[i]);      // deg >= 1 always (self loops)
}

// ---------------- WMMA GEMM: Out[M x NTOT] = A[M x KTOT] @ W[KTOT x NREAL] --
// 128-row blocks, 8 waves; each wave computes 16 rows x NTOT cols.
// fp32 WMMA 16x16x4.
//   A tile: row-major, +4 dword row pad (conflict-free ds_load_b64 pairs).
//   W tile: K-interleaved Wp[k/4][n][k%4] so each lane's {k+kb, k+kb+1} pair
//           is one aligned 8-byte LDS load -> direct v2f B operand, no movs.

template <int KTOT, int NTOT, int NREAL>
__global__ __launch_bounds__(256) void k_gemm_wmma(const float* __restrict__ A, int lda,
                                                   const float* __restrict__ W, int ldw,
                                                   float* __restrict__ Out, int M) {
    constexpr int BM = 128;
    constexpr int KC = 64;
    constexpr int NT = NTOT / 16;
    __shared__ float Xs[BM][KC + 4];
    __shared__ float Wp[KC / 4][NTOT][4];

    const int tid  = threadIdx.x;
    const int lane = tid & 31;
    const int wave = tid >> 5;
    const int m0   = blockIdx.x * BM;

    const int mlo = lane & 15;          // M (for A) / N (for B) sub-index
    const int kb  = (lane >> 4) << 1;   // K pair base: lanes 0-15 -> {0,1}, 16-31 -> {2,3}

    v8f acc[NT] = {};

    for (int kc = 0; kc < KTOT; kc += KC) {
        // stage A tile (BM x KC) as float4; prefetch next chunk while at it
        for (int i = tid; i < BM * (KC / 4); i += 256) {
            int r  = i / (KC / 4);
            int c4 = (i % (KC / 4)) * 4;
            v4f v = {};
            int gm = m0 + r;
            if (gm < M) {
                const float* p = A + (size_t)gm * lda + kc + c4;
                v = *(const v4f*)p;
                if constexpr (KTOT > KC) {
                    if (kc + KC < KTOT) __builtin_prefetch(p + KC, 0, 1);
                }
            }
            *(v4f*)(&Xs[r][c4]) = v;
        }
        // stage W tile (KC x NTOT) K-interleaved, zero-fill cols >= NREAL
        for (int i = tid; i < KC * NTOT; i += 256) {
            int r = i / NTOT, c = i % NTOT;
            float wv = 0.0f;
            if (c < NREAL) wv = W[(size_t)(kc + r) * ldw + c];
            Wp[r >> 2][c][r & 3] = wv;
        }
        __syncthreads();

        const int mrow = wave * 16 + mlo;
#pragma unroll
        for (int k = 0; k < KC; k += 4) {
            v2f a = *(const v2f*)(&Xs[mrow][k + kb]);
#pragma unroll
            for (int t = 0; t < NT; ++t) {
                v2f b = *(const v2f*)(&Wp[k >> 2][t * 16 + mlo][kb]);
                acc[t] = __builtin_amdgcn_wmma_f32_16x16x4_f32(
                    false, a, false, b, (short)0, acc[t], false, false);
            }
        }
        __syncthreads();
    }

    // C/D layout: VGPR r -> (lanes 0-15: M=r, N=lane), (lanes 16-31: M=r+8, N=lane-16)
    const int mbase = m0 + wave * 16 + ((lane >> 4) << 3);
#pragma unroll
    for (int r = 0; r < 8; ++r) {
        int gm = mbase + r;
        if (gm < M) {
#pragma unroll
            for (int t = 0; t < NT; ++t)
                Out[(size_t)gm * NTOT + t * 16 + mlo] = acc[t][r];
        }
    }
}

// ---------------- aggregation ----------------

// agg[i,:] = dense[i,:] * dinv[i]^2   (self-loop term; also zero-initializes agg)
template <int D>
__global__ __launch_bounds__(256) void k_self_init(const float* __restrict__ dense,
                                                   const float* __restrict__ dinv,
                                                   float* __restrict__ agg) {
    size_t t = (size_t)blockIdx.x * 256 + threadIdx.x;
    if (t >= (size_t)NNODES * D) return;
    int i = (int)(t / D);
    float w = dinv[i];
    agg[t] = dense[t] * (w * w);
}

// agg[dst,:] += dense[src,:] * dinv[src] * dinv[dst]  over all real edges.
// One wave32 per edge: edge index is wave-uniform (readfirstlane -> scalar
// loads of src/dst/norm), lanes do coalesced 128B gather + atomic scatter.
template <int D>
__global__ __launch_bounds__(256) void k_scatter_add(const int* __restrict__ src,
                                                     const int* __restrict__ dst,
                                                     const float* __restrict__ dinv,
                                                     const float* __restrict__ dense,
                                                     float* __restrict__ agg, int E) {
    int lane = threadIdx.x & 31;
    int e = __builtin_amdgcn_readfirstlane(blockIdx.x * 8 + (threadIdx.x >> 5));
    if (e >= E) return;
    int s = src[e];
    int d = dst[e];
    float nrm = dinv[s] * dinv[d];
    const float* row = dense + (size_t)s * D;
    float*       acc = agg   + (size_t)d * D;
#pragma unroll
    for (int f = lane; f < D; f += 32)
        atomicAdd(&acc[f], row[f] * nrm);
}

__global__ __launch_bounds__(256) void k_bias_relu64(float* __restrict__ h,
                                                     const float* __restrict__ b) {
    size_t t = (size_t)blockIdx.x * 256 + threadIdx.x;
    if (t >= (size_t)NNODES * 64) return;
    float v = h[t] + b[t & 63];
    h[t] = v > 0.0f ? v : 0.0f;
}

// ---------------- log_softmax over 47 classes, one wave32 per row ----------

__global__ __launch_bounds__(256) void k_logsoftmax47(const float* __restrict__ agg2,
                                                      const float* __restrict__ b2,
                                                      float* __restrict__ out) {
    int lane = threadIdx.x & 31;
    int row  = blockIdx.x * 8 + (threadIdx.x >> 5);
    if (row >= NNODES) return;
    const float* r = agg2 + (size_t)row * 48;

    const float NEG = -3.0e38f;
    float v0 = (lane < 47)      ? r[lane] + b2[lane]           : NEG;
    float v1 = (lane + 32 < 47) ? r[lane + 32] + b2[lane + 32] : NEG;

    float m = fmaxf(v0, v1);
#pragma unroll
    for (int off = 16; off; off >>= 1) m = fmaxf(m, __shfl_xor(m, off, 32));

    float s = 0.0f;
    if (lane < 47)      s += __expf(v0 - m);
    if (lane + 32 < 47) s += __expf(v1 - m);
#pragma unroll
    for (int off = 16; off; off >>= 1) s += __shfl_xor(s, off, 32);

    float lse = __logf(s) + m;
    if (lane < 47)      out[(size_t)row * 47 + lane]      = v0 - lse;
    if (lane + 32 < 47) out[(size_t)row * 47 + lane + 32] = v1 - lse;
}

// ---------------- launcher ----------------

extern "C" void kernel_launch(void* const* d_in, const int* in_sizes, int n_in,
                              void* d_out, int out_size, void* d_ws, size_t ws_size,
                              hipStream_t stream) {
    const float* x  = (const float*)d_in[0];   // [100000, 256]
    const int*   ei = (const int*)d_in[1];     // [2, E]
    const float* W1 = (const float*)d_in[2];   // [256, 64]
    const float* b1 = (const float*)d_in[3];   // [64]
    const float* W2 = (const float*)d_in[4];   // [64, 47]
    const float* b2 = (const float*)d_in[5];   // [47]
    float* out = (float*)d_out;                // [100000, 47]

    const int E = in_sizes[1] / 2;
    const int* src = ei;
    const int* dst = ei + E;

    // workspace layout (floats): dinv | H(=G reuse) | agg1(->h1) | agg2
    float* ws   = (float*)d_ws;
    float* dinv = ws;                                    // NNODES (deg in-place)
    float* H    = ws + 100352;                           // NNODES*64
    float* agg1 = H + (size_t)NNODES * 64;               // NNODES*64
    float* agg2 = agg1 + (size_t)NNODES * 64;            // NNODES*48
    float* G    = H;                                     // reuse after layer 1

    const int gN    = (NNODES + 255) / 256;
    const int gE    = (E + 255) / 256;
    const int gEw   = (E + 7) / 8;                       // wave-per-edge kernels
    const int gGemm = (NNODES + 127) / 128;

    // --- normalization ---
    k_deg_init<<<gN, 256, 0, stream>>>(dinv);
    k_deg_accum<<<gE, 256, 0, stream>>>(dst, E, dinv);
    k_deg_to_dinv<<<gN, 256, 0, stream>>>(dinv);

    // --- layer 1 ---
    k_gemm_wmma<256, 64, 64><<<gGemm, 256, 0, stream>>>(x, 256, W1, 64, H, NNODES);
    {
        int g = (int)(((size_t)NNODES * 64 + 255) / 256);
        k_self_init<64><<<g, 256, 0, stream>>>(H, dinv, agg1);
    }
    k_scatter_add<64><<<gEw, 256, 0, stream>>>(src, dst, dinv, H, agg1, E);
    {
        int g = (int)(((size_t)NNODES * 64 + 255) / 256);
        k_bias_relu64<<<g, 256, 0, stream>>>(agg1, b1);  // agg1 is now h1
    }

    // --- layer 2 (N padded 47 -> 48) ---
    k_gemm_wmma<64, 48, 47><<<gGemm, 256, 0, stream>>>(agg1, 64, W2, 47, G, NNODES);
    {
        int g = (int)(((size_t)NNODES * 48 + 255) / 256);
        k_self_init<48><<<g, 256, 0, stream>>>(G, dinv, agg2);
    }
    k_scatter_add<48><<<gEw, 256, 0, stream>>>(src, dst, dinv, G, agg2, E);
    k_logsoftmax47<<<(NNODES + 7) / 8, 256, 0, stream>>>(agg2, b2, out);
}